// SkeletonGNN_87780541596400
// MI455X (gfx1250) — compile-verified
//
#include <hip/hip_runtime.h>

typedef __attribute__((ext_vector_type(2))) float v2f;
typedef __attribute__((ext_vector_type(8))) float v8f;

#define NCOL 128   // DH == DOUT == 128

// ---------------------------------------------------------------------------
// Degree / dinv kernels (shared graph structure, computed once)
// ---------------------------------------------------------------------------
__global__ void init_deg_kernel(float* __restrict__ deg, int n) {
    int i = blockIdx.x * 256 + threadIdx.x;
    if (i < n) deg[i] = 1.0f;   // self-loop
}

__global__ void deg_accum_kernel(const int* __restrict__ dst, float* __restrict__ deg, int e) {
    int i = blockIdx.x * 256 + threadIdx.x;
    if (i < e) atomicAdd(&deg[dst[i]], 1.0f);
}

__global__ void dinv_kernel(const float* __restrict__ deg, float* __restrict__ dinv, int n) {
    int i = blockIdx.x * 256 + threadIdx.x;
    if (i < n) dinv[i] = rsqrtf(deg[i]);   // deg >= 1 always (self-loop)
}

// ---------------------------------------------------------------------------
// GEMM: hws[r,:] = (A[r,:] @ W) * dinv[r], also acc = hws (self-loop init)
// fp32 WMMA 16x16x4. Block = 256 threads = 8 waves; each wave owns 16 rows
// and sweeps the 8 column tiles of the 128-wide output. W staged in LDS.
// ---------------------------------------------------------------------------
template <int K>
__global__ __launch_bounds__(256)
void gemm_scale_kernel(const float* __restrict__ A,
                       const float* __restrict__ W,
                       const float* __restrict__ dinv,
                       float* __restrict__ hws,
                       float* __restrict__ acc,
                       int nrows)
{
    __shared__ float lds_w[K * NCOL];
    const int tid = threadIdx.x;
    for (int i = tid; i < K * NCOL; i += 256) lds_w[i] = W[i];
    __syncthreads();

    const int wave = tid >> 5;
    const int lane = tid & 31;
    const int hi   = lane >> 4;   // 0: lanes 0-15, 1: lanes 16-31
    const int lm   = lane & 15;
    const int row0 = blockIdx.x * 128 + wave * 16;

    // Preload all A fragments for this wave's 16-row tile.
    // 32-bit A 16x4 layout: lanes 0-15 hold M=0..15; VGPR0 = K {0|2}, VGPR1 = K {1|3}.
    v2f afrag[K / 4];
    {
        int r = row0 + lm;
        if (r > nrows - 1) r = nrows - 1;           // clamp tail (stores guarded later)
        const float* ap = A + (size_t)r * K + hi * 2;
        #pragma unroll
        for (int s = 0; s < K / 4; ++s)
            afrag[s] = *(const v2f*)(ap + s * 4);
    }

    // dinv for the 8 output rows this lane writes (M = v + hi*8)
    float dv[8];
    #pragma unroll
    for (int v = 0; v < 8; ++v) {
        int r = row0 + v + hi * 8;
        if (r > nrows - 1) r = nrows - 1;
        dv[v] = dinv[r];
    }

    #pragma unroll
    for (int nt = 0; nt < 8; ++nt) {
        v8f c = {0.f, 0.f, 0.f, 0.f, 0.f, 0.f, 0.f, 0.f};
        const int bcol = nt * 16 + lm;
        #pragma unroll
        for (int s = 0; s < K / 4; ++s) {
            // 32-bit B 4x16: lanes hold N=0..15; VGPR0 = K {0|2}, VGPR1 = K {1|3}
            const int kb = s * 4 + hi * 2;
            v2f b;
            b.x = lds_w[kb * NCOL + bcol];
            b.y = lds_w[(kb + 1) * NCOL + bcol];
            c = __builtin_amdgcn_wmma_f32_16x16x4_f32(
                    false, afrag[s], false, b, (short)0, c, false, false);
        }
        // C/D layout: VGPR v -> M = v (+8 for lanes 16-31), N = lane&15
        #pragma unroll
        for (int v = 0; v < 8; ++v) {
            const int r = row0 + v + hi * 8;
            if (r < nrows) {
                const float val = c[v] * dv[v];
                const size_t idx = (size_t)r * NCOL + bcol;
                hws[idx] = val;
                acc[idx] = val;
            }
        }
    }
}

// ---------------------------------------------------------------------------
// Edge scatter: acc[dst[e],:] += hws[src[e],:]  (32 lanes/edge, float4/lane)
// ---------------------------------------------------------------------------
__global__ __launch_bounds__(256)
void scatter_kernel(const int* __restrict__ src, const int* __restrict__ dst,
                    const float* __restrict__ hws, float* __restrict__ acc, int e)
{
    const unsigned t = blockIdx.x * 256u + threadIdx.x;
    const unsigned edge = t >> 5;
    const int lane = t & 31;
    if (edge >= (unsigned)e) return;
    const int s = src[edge];
    const int d = dst[edge];
    const float4 v = *(const float4*)(hws + (size_t)s * NCOL + lane * 4);
    float* p = acc + (size_t)d * NCOL + lane * 4;
    atomicAdd(p + 0, v.x);
    atomicAdd(p + 1, v.y);
    atomicAdd(p + 2, v.z);
    atomicAdd(p + 3, v.w);
}

// ---------------------------------------------------------------------------
// Epilogue: out = relu(LayerNorm(acc*dinv[row] + bias; gamma, beta))
// One wave (32 lanes x 4 cols) per 128-wide row.
// ---------------------------------------------------------------------------
__global__ __launch_bounds__(256)
void post_ln_relu_kernel(const float* __restrict__ acc, const float* __restrict__ dinv,
                         const float* __restrict__ bias, const float* __restrict__ gamma,
                         const float* __restrict__ beta, float* __restrict__ out, int n)
{
    const int wave = threadIdx.x >> 5;
    const int lane = threadIdx.x & 31;
    const int row = blockIdx.x * 8 + wave;
    if (row >= n) return;                  // whole wave exits together

    const float dv = dinv[row];
    const int c0 = lane * 4;
    float4 v = *(const float4*)(acc + (size_t)row * NCOL + c0);
    const float4 b = *(const float4*)(bias + c0);
    v.x = v.x * dv + b.x;
    v.y = v.y * dv + b.y;
    v.z = v.z * dv + b.z;
    v.w = v.w * dv + b.w;

    float sum = v.x + v.y + v.z + v.w;
    float ssq = v.x * v.x + v.y * v.y + v.z * v.z + v.w * v.w;
    #pragma unroll
    for (int m = 16; m >= 1; m >>= 1) {
        sum += __shfl_xor(sum, m, 32);
        ssq += __shfl_xor(ssq, m, 32);
    }
    const float mu   = sum * (1.0f / 128.0f);
    const float var  = ssq * (1.0f / 128.0f) - mu * mu;
    const float rstd = rsqrtf(var + 1e-5f);

    const float4 g  = *(const float4*)(gamma + c0);
    const float4 bt = *(const float4*)(beta + c0);
    float4 o;
    o.x = fmaxf((v.x - mu) * rstd * g.x + bt.x, 0.0f);
    o.y = fmaxf((v.y - mu) * rstd * g.y + bt.y, 0.0f);
    o.z = fmaxf((v.z - mu) * rstd * g.z + bt.z, 0.0f);
    o.w = fmaxf((v.w - mu) * rstd * g.w + bt.w, 0.0f);
    *(float4*)(out + (size_t)row * NCOL + c0) = o;
}

// ---------------------------------------------------------------------------
extern "C" void kernel_launch(void* const* d_in, const int* in_sizes, int n_in,
                              void* d_out, int out_size, void* d_ws, size_t ws_size,
                              hipStream_t stream)
{
    const int DH  = in_sizes[3];                 // b1 -> 128
    const int DIN = in_sizes[2] / DH;            // W1 -> 64
    const int N   = in_sizes[0] / DIN;           // 100000
    const int E   = in_sizes[1] / 2;             // 1600000

    const float* x   = (const float*)d_in[0];
    const int*   ei  = (const int*)d_in[1];
    const float* W1  = (const float*)d_in[2];
    const float* b1  = (const float*)d_in[3];
    const float* g1  = (const float*)d_in[4];
    const float* bt1 = (const float*)d_in[5];
    const float* W2  = (const float*)d_in[6];
    const float* b2  = (const float*)d_in[7];
    const float* g2  = (const float*)d_in[8];
    const float* bt2 = (const float*)d_in[9];

    const int* src = ei;          // edge_index[0]
    const int* dst = ei + E;      // edge_index[1]

    // Workspace layout (all fp32)
    float* deg  = (float*)d_ws;
    float* dinv = deg + N;
    float* hws  = dinv + N;
    float* acc  = hws + (size_t)N * NCOL;
    float* h1   = acc + (size_t)N * NCOL;

    const int nb_n = (N + 255) / 256;
    const int nb_e = (E + 255) / 256;
    const int nb_g = (N + 127) / 128;
    const int nb_s = (int)(((size_t)E * 32 + 255) / 256);
    const int nb_p = (N + 7) / 8;

    // Graph structure (shared by both layers)
    init_deg_kernel<<<nb_n, 256, 0, stream>>>(deg, N);
    deg_accum_kernel<<<nb_e, 256, 0, stream>>>(dst, deg, E);
    dinv_kernel<<<nb_n, 256, 0, stream>>>(deg, dinv, N);

    // Layer 1: x[N,64] @ W1 -> LN -> ReLU -> h1
    gemm_scale_kernel<64><<<nb_g, 256, 0, stream>>>(x, W1, dinv, hws, acc, N);
    scatter_kernel<<<nb_s, 256, 0, stream>>>(src, dst, hws, acc, E);
    post_ln_relu_kernel<<<nb_p, 256, 0, stream>>>(acc, dinv, b1, g1, bt1, h1, N);

    // Layer 2: h1[N,128] @ W2 -> LN -> ReLU -> out
    gemm_scale_kernel<128><<<nb_g, 256, 0, stream>>>(h1, W2, dinv, hws, acc, N);
    scatter_kernel<<<nb_s, 256, 0, stream>>>(src, dst, hws, acc, E);
    post_ln_relu_kernel<<<nb_p, 256, 0, stream>>>(acc, dinv, b2, g2, bt2, (float*)d_out, N);
}